// LightGCN_joint_50044958933434
// MI455X (gfx1250) — compile-verified
//
#include <hip/hip_runtime.h>

#define NUM_USERS 100000
#define NUM_ITEMS 50000
#define NTOT      (NUM_USERS + NUM_ITEMS)
#define DIM       64
#define NLAYERS   3

// ---------------------------------------------------------------------------
// init: cur = concat(user_emb, item_emb); acc(d_out) = same   (float4 path)
// ---------------------------------------------------------------------------
__global__ __launch_bounds__(256) void lgcn_init(const float4* __restrict__ ue,
                                                 const float4* __restrict__ ie,
                                                 float4* __restrict__ cur,
                                                 float4* __restrict__ acc,
                                                 int n4user, int n4tot) {
    int i = blockIdx.x * blockDim.x + threadIdx.x;
    if (i < n4tot) {
        float4 v = (i < n4user) ? ue[i] : ie[i - n4user];
        cur[i] = v;
        acc[i] = v;
    }
}

__global__ __launch_bounds__(256) void lgcn_zero(float4* __restrict__ p, int n4) {
    int i = blockIdx.x * blockDim.x + threadIdx.x;
    if (i < n4) p[i] = make_float4(0.f, 0.f, 0.f, 0.f);
}

// acc = (acc + nxt) * scale   (scale = 1 for inner layers, 0.25 fused on last)
__global__ __launch_bounds__(256) void lgcn_accum(float4* __restrict__ acc,
                                                  const float4* __restrict__ nxt,
                                                  float scale, int n4) {
    int i = blockIdx.x * blockDim.x + threadIdx.x;
    if (i < n4) {
        float4 a = acc[i];
        float4 b = nxt[i];
        a.x = (a.x + b.x) * scale;
        a.y = (a.y + b.y) * scale;
        a.z = (a.z + b.z) * scale;
        a.w = (a.w + b.w) * scale;
        acc[i] = a;
    }
}

// ---------------------------------------------------------------------------
// COO SpMM scatter:  nxt[row[e]] += vals[e] * cur[col[e]]
// One wave32 handles 2 edges per iteration: 16 lanes x float4 = one 256B row.
// Gather goes global -> LDS via async b128 loads, double-buffered (ASYNCcnt),
// scatter via hardware fp32 atomics (output lives in L2).
// ---------------------------------------------------------------------------
__global__ __launch_bounds__(256) void lgcn_spmm(const float* __restrict__ cur,
                                                 float* __restrict__ nxt,
                                                 const float* __restrict__ vals,
                                                 const int* __restrict__ rowi,
                                                 const int* __restrict__ coli,
                                                 int nnz) {
    // 8 waves/block * 2 stages * 128 floats (2 edges x 64) = 8 KB
    __shared__ __align__(16) float lds[8 * 2 * 128];

    const int lane = threadIdx.x & 31;
    const int wave = threadIdx.x >> 5;
    const int sub  = lane >> 4;   // which edge of the pair
    const int l16  = lane & 15;   // 16B chunk within the 256B row

    const int gw     = blockIdx.x * 8 + wave;     // global wave id
    const int nw     = gridDim.x * 8;             // total waves
    const int stride = nw * 2;                    // edges per sweep

    float* wbase = &lds[wave * 256];

    auto issue = [&](int b, int stage) {
        int e = b + sub;
        int ec = (e < nnz) ? e : (nnz - 1);       // clamp tail lanes (harmless)
        int c  = coli[ec];
        const float* gp = cur + (size_t)c * DIM + l16 * 4;
        unsigned long long ga = (unsigned long long)(uintptr_t)gp;
        unsigned la = (unsigned)(uintptr_t)(wbase + stage * 128 + lane * 4);
        asm volatile("global_load_async_to_lds_b128 %0, %1, off"
                     :: "v"(la), "v"(ga) : "memory");
    };

    int eb = gw * 2;
    if (eb < nnz) issue(eb, 0);                   // prologue prefetch
    int stage = 0;

    for (; eb < nnz; eb += stride) {
        int nb = eb + stride;                     // wave-uniform
        if (nb < nnz) {
            issue(nb, stage ^ 1);                 // prefetch next batch
            asm volatile("s_wait_asynccnt 0x1" ::: "memory"); // current done
        } else {
            asm volatile("s_wait_asynccnt 0x0" ::: "memory");
        }

        int e = eb + sub;
        if (e < nnz) {
            float v = vals[e];
            int   r = rowi[e];
            const float4 m =
                *reinterpret_cast<const float4*>(wbase + stage * 128 + lane * 4);
            float* dst = nxt + (size_t)r * DIM + l16 * 4;
            unsafeAtomicAdd(dst + 0, m.x * v);
            unsafeAtomicAdd(dst + 1, m.y * v);
            unsafeAtomicAdd(dst + 2, m.z * v);
            unsafeAtomicAdd(dst + 3, m.w * v);
        }
        stage ^= 1;
    }
}

// ---------------------------------------------------------------------------
extern "C" void kernel_launch(void* const* d_in, const int* in_sizes, int n_in,
                              void* d_out, int out_size, void* d_ws, size_t ws_size,
                              hipStream_t stream) {
    (void)n_in; (void)out_size; (void)ws_size;

    const float* ue   = (const float*)d_in[0];
    const float* ie   = (const float*)d_in[1];
    const float* vals = (const float*)d_in[2];
    const int*   row  = (const int*)d_in[3];
    const int*   col  = (const int*)d_in[4];
    float*       out  = (float*)d_out;

    float* ws0 = (float*)d_ws;                        // ping
    float* ws1 = ws0 + (size_t)NTOT * DIM;            // pong (2 x 38.4 MB)

    const int nnz    = in_sizes[2];
    const int n4tot  = NTOT * DIM / 4;                // 2,400,000 float4
    const int n4user = NUM_USERS * DIM / 4;
    const int eblk   = (n4tot + 255) / 256;

    lgcn_init<<<eblk, 256, 0, stream>>>((const float4*)ue, (const float4*)ie,
                                        (float4*)ws0, (float4*)out, n4user, n4tot);

    float* cur = ws0;
    float* nxt = ws1;
    for (int l = 0; l < NLAYERS; ++l) {
        lgcn_zero<<<eblk, 256, 0, stream>>>((float4*)nxt, n4tot);
        lgcn_spmm<<<3072, 256, 0, stream>>>(cur, nxt, vals, row, col, nnz);
        lgcn_accum<<<eblk, 256, 0, stream>>>((float4*)out, (const float4*)nxt,
                                             (l == NLAYERS - 1) ? 0.25f : 1.0f, n4tot);
        float* t = cur; cur = nxt; nxt = t;
    }
}